// Refiner_14989435863198
// MI455X (gfx1250) — compile-verified
//
#include <hip/hip_runtime.h>
#include <hip/hip_bf16.h>
#include <stdint.h>

// ---------------------------------------------------------------------------
// Beamforming refiner: B=32768 independent 2-param optimizers, T=10 steps,
// K=256 elements. Analytic gradient -> single-float wave reduction per step.
// One wave32 per batch element; z kept register-resident for all T steps.
// CDNA5 paths: global_load_async_to_lds_b128 + s_wait_asynccnt staging,
// ds_load_b128 reads, V_WMMA_F32_16X16X4_F32 ones-matrix wave reduction,
// v_sin/v_cos/v_exp/v_rcp trans ops.
// ---------------------------------------------------------------------------

namespace {

constexpr int KLEN = 256;
constexpr int WPB  = 8;          // waves per block
constexpr int TPB  = WPB * 32;   // 256 threads = 8 wave32

constexpr float PI_F      = 3.14159265358979323846f;
constexpr float TWO_PI_F  = 6.28318530717958647692f;
constexpr float INV_2PI_F = 0.15915494309189533577f;
// dphase/dtheta_deg factor: 2*pi*SPATIAL*(pi/180), SPATIAL=0.5 -> pi^2/180
constexpr float CT_F = 0.05483113556160755f;
// dphase/dr factor: -2*pi*RANGE_C, RANGE_C = 2e-4
constexpr float CR_F = -1.2566370614359172e-3f;
constexpr float C_NORM = 0.0625f;   // 1/sqrt(256)

typedef __attribute__((ext_vector_type(2))) float v2f;
typedef __attribute__((ext_vector_type(8))) float v8f;

__device__ __forceinline__ float frcp(float x) { return __builtin_amdgcn_rcpf(x); }
__device__ __forceinline__ float clampf(float x, float lo, float hi) {
  return fminf(fmaxf(x, lo), hi);
}
__device__ __forceinline__ float tanh_fast(float x) {
  // tanh(x) = 1 - 2/(exp(2x)+1); exact limits at +-inf
  float e = __expf(2.0f * x);
  return 1.0f - 2.0f * frcp(e + 1.0f);
}
__device__ __forceinline__ float softplus_f(float x) {
  return (x > 15.0f) ? x : log1pf(__expf(x));
}
__device__ __forceinline__ float atanh_clipped(float y) {
  y = clampf(y, -1.0f + 1e-6f, 1.0f - 1e-6f);
  return 0.5f * __logf((1.0f + y) / (1.0f - y));
}
__device__ __forceinline__ float finite_or(float x, float alt) {
  return isfinite(x) ? x : alt;
}

// CDNA5 async global->LDS copy (tracked by ASYNCcnt). First operand is the
// VGPR holding the wave-relative LDS byte offset, second the 64-bit global
// address. Generic LDS pointers carry the LDS offset in their low 32 bits.
__device__ __forceinline__ void async_g2lds_b128(uint32_t lds_off, const void* gptr) {
  asm volatile("global_load_async_to_lds_b128 %0, %1, off"
               :: "v"(lds_off), "v"(gptr)
               : "memory");
}
__device__ __forceinline__ void wait_asynccnt0() {
  asm volatile("s_wait_asynccnt 0" ::: "memory");
}

// Wave32 all-lane sum. Preferred path: one V_WMMA_F32_16X16X4_F32 against an
// all-ones B matrix.  A-operand layout (ISA 7.12.2, 32-bit A 16x4):
//   VGPR0: lanes 0-15 -> A[m][0] (m = lane), lanes 16-31 -> A[m][2] (m = lane-16)
//   VGPR1: K=1 / K=3  (we pass zeros)
// With B = ones (layout-independent): D[m][n] = x[m] + x[m+16].
// D layout: VGPR j, lanes 0-15 -> D[j][n]; lanes 16-31 -> D[j+8][n].
// Sum of the 8 D VGPRs gives half-sum per lane-half; one xor-16 swap adds the
// other half -> full sum uniform across the wave.  Requires EXEC == all ones
// (guaranteed: the only branch guarding this code is wave-uniform).
__device__ __forceinline__ float wave32_reduce_add(float v) {
#if __has_builtin(__builtin_amdgcn_wmma_f32_16x16x4_f32)
  v2f a;
  a.x = v;
  a.y = 0.0f;
  v2f ones;
  ones.x = 1.0f;
  ones.y = 1.0f;
  v8f c = {};
  v8f d = __builtin_amdgcn_wmma_f32_16x16x4_f32(
      /*neg_a=*/false, a, /*neg_b=*/false, ones,
      /*c_mod=*/(short)0, c, /*reuse_a=*/false, /*reuse_b=*/false);
  float s = ((d[0] + d[1]) + (d[2] + d[3])) + ((d[4] + d[5]) + (d[6] + d[7]));
  s += __shfl_xor(s, 16, 32);
  return s;
#else
#pragma unroll
  for (int m = 16; m >= 1; m >>= 1) {
    v += __shfl_xor(v, m, 32);
  }
  return v;
#endif
}

__global__ __launch_bounds__(TPB) void beam_refine_kernel(
    const float* __restrict__ z_real,
    const float* __restrict__ z_imag,
    const float* __restrict__ theta0_deg,
    const float* __restrict__ r0_m,
    const float* __restrict__ a_th_raw,
    const float* __restrict__ a_r_raw,
    const float* __restrict__ l_th_raw,
    const float* __restrict__ l_r_raw,
    float* __restrict__ out,   // (B,2) [theta, r]
    int B, int T)
{
  __shared__ __align__(16) float smem[WPB * 2 * KLEN];  // 16 KB / block

  const int wave = threadIdx.x >> 5;
  const int lane = threadIdx.x & 31;
  const int b = blockIdx.x * WPB + wave;
  if (b >= B) return;   // wave-uniform guard (EXEC stays all-ones inside)

  float* lre = smem + wave * (2 * KLEN);
  float* lim = lre + KLEN;

  // ---- Stage this wave's 2 KB of z via async DMA to LDS (wave-private) ----
  {
    const char* gre = (const char*)(z_real + (size_t)b * KLEN);
    const char* gim = (const char*)(z_imag + (size_t)b * KLEN);
    const uint32_t ore = (uint32_t)(uintptr_t)lre;
    const uint32_t oim = (uint32_t)(uintptr_t)lim;
    const uint32_t lo  = (uint32_t)(lane * 16);
    async_g2lds_b128(ore + lo,       gre + lo);
    async_g2lds_b128(ore + 512 + lo, gre + 512 + lo);
    async_g2lds_b128(oim + lo,       gim + lo);
    async_g2lds_b128(oim + 512 + lo, gim + 512 + lo);
    wait_asynccnt0();   // own-wave asynccnt only; no barrier needed
  }

  // ---- Pull lane's 8 contiguous k values into registers, pre-scaled ----
  // lane owns k in [8*lane, 8*lane+8)
  float zc_re[8], zc_im[8];  // 0.125 * z  (beta * z / sqrt(K))
  float z2_re[8], z2_im[8];  // 2 * z      (beta * z)
  {
    const float4* pre = (const float4*)(lre + lane * 8);
    const float4* pim = (const float4*)(lim + lane * 8);
    float4 A0 = pre[0], A1 = pre[1];
    float4 B0 = pim[0], B1 = pim[1];
    float tre[8] = {A0.x, A0.y, A0.z, A0.w, A1.x, A1.y, A1.z, A1.w};
    float tim[8] = {B0.x, B0.y, B0.z, B0.w, B1.x, B1.y, B1.z, B1.w};
#pragma unroll
    for (int j = 0; j < 8; ++j) {
      zc_re[j] = 0.125f * tre[j];
      z2_re[j] = 2.0f   * tre[j];
      zc_im[j] = 0.125f * tim[j];
      z2_im[j] = 2.0f   * tim[j];
    }
  }
  const float k0 = (float)(lane * 8);

  // ---- Initial u from (theta0, r0) ----
  float u0, u1;
  {
    float th0 = theta0_deg[b];
    float rr0 = r0_m[b];
    u0 = atanh_clipped(th0 * (1.0f / 60.0f));
    float s = clampf(rr0 * (1.0f / 2000.0f), 1e-6f, 1.0f - 1e-6f);
    u1 = atanh_clipped(2.0f * s - 1.0f);
  }

  // ---- T serial refinement steps, all data in registers ----
  for (int t = 0; t < T; ++t) {
    float a_th = finite_or(clampf(softplus_f(a_th_raw[t]), 1e-6f, 0.2f), 0.01f);
    float a_r  = finite_or(clampf(softplus_f(a_r_raw[t]),  1e-6f, 0.2f), 0.01f);
    float l_th = finite_or(clampf(softplus_f(l_th_raw[t]), 1e-6f, 1.0f), 0.001f);
    float l_r  = finite_or(clampf(softplus_f(l_r_raw[t]),  1e-6f, 1.0f), 0.001f);

    float t0 = tanh_fast(u0);
    float t1 = tanh_fast(u1);
    float th_rad = t0 * (60.0f * PI_F / 180.0f);      // theta_deg * pi/180
    float sin_t = __sinf(th_rad);
    float cos_t = __cosf(th_rad);
    float r = 1000.0f * (t1 + 1.0f);

    // phi = 2*pi*(SPATIAL*sin(th) - RANGE_C*r); reduce mod 2*pi (k integer
    // => sin/cos(k*phi) unchanged, keeps hw trans-ops in accurate range)
    float phi = PI_F * sin_t - 1.2566370614359172e-3f * r;
    phi -= TWO_PI_F * rintf(phi * INV_2PI_F);

    float S = 0.0f;
#pragma unroll
    for (int j = 0; j < 8; ++j) {
      float kf = k0 + (float)j;
      float p = kf * phi;
      p -= TWO_PI_F * rintf(p * INV_2PI_F);           // |p| <= pi
      float sn = __sinf(p);
      float cs = __cosf(p);
      // w = beta*z*sigmoid(-beta*z*a) = 2z / (1 + exp(0.125*z*trig));
      // share one reciprocal between both sigmoids:
      //   inv = 1/((1+e_re)(1+e_im)); 1/(1+e_re) = (1+e_im)*inv; etc.
      float are = 1.0f + __expf(zc_re[j] * cs);
      float aim = 1.0f + __expf(zc_im[j] * sn);
      float inv = frcp(are * aim);
      float wre = z2_re[j] * (aim * inv);
      float wim = z2_im[j] * (are * inv);
      float m = fmaf(cs, wim, -sn * wre);             // dll/dp_k / c
      S = fmaf(kf, m, S);
    }
    S = wave32_reduce_add(S);                         // uniform across wave

    float G = C_NORM * S;                             // sum_k k * dll/dp_k
    float gth = -(G * CT_F * cos_t) * (60.0f * (1.0f - t0 * t0));
    float gr  = -(G * CR_F)         * (1000.0f * (1.0f - t1 * t1));
    gth = finite_or(gth, 0.0f);
    gr  = finite_or(gr,  0.0f);

    float den_th = fmaxf(fmaxf(fabsf(gth), 1e-6f) + l_th, 1e-6f);
    float den_r  = fmaxf(fmaxf(fabsf(gr),  1e-6f) + l_r,  1e-6f);
    float st = clampf(a_th * gth * frcp(den_th), -0.1f, 0.1f);
    float sr = clampf(a_r  * gr  * frcp(den_r),  -0.1f, 0.1f);
    u0 -= st;
    u1 -= sr;
  }

  if (lane == 0) {
    float th = 60.0f * tanh_fast(u0);
    float rr = 1000.0f * (tanh_fast(u1) + 1.0f);
    ((float2*)out)[b] = make_float2(th, rr);
  }
}

} // namespace

extern "C" void kernel_launch(void* const* d_in, const int* in_sizes, int n_in,
                              void* d_out, int out_size, void* d_ws, size_t ws_size,
                              hipStream_t stream) {
  const float* z_real = (const float*)d_in[0];
  const float* z_imag = (const float*)d_in[1];
  const float* theta0 = (const float*)d_in[2];
  const float* r0m    = (const float*)d_in[3];
  const float* ath    = (const float*)d_in[4];
  const float* ar     = (const float*)d_in[5];
  const float* lth    = (const float*)d_in[6];
  const float* lr     = (const float*)d_in[7];

  const int B = in_sizes[2];   // theta0 length
  const int T = in_sizes[4];   // alpha_theta_raw length

  const int grid = (B + WPB - 1) / WPB;
  beam_refine_kernel<<<grid, TPB, 0, stream>>>(
      z_real, z_imag, theta0, r0m, ath, ar, lth, lr,
      (float*)d_out, B, T);
}